// RelGraphConv_78005196030450
// MI455X (gfx1250) — compile-verified
//
#include <hip/hip_runtime.h>

typedef __attribute__((ext_vector_type(2))) float v2f;
typedef __attribute__((ext_vector_type(8))) float v8f;
typedef __attribute__((ext_vector_type(4))) int   v4i;
typedef __attribute__((address_space(1))) v4i*    gv4i_p;   // global (prints as __device__*)
typedef __attribute__((address_space(3))) v4i*    lv4i_p;   // LDS

#define IN_FEAT   64
#define OUT_FEAT  64
#define NUM_BASES 8
#define KBASIS    (NUM_BASES * IN_FEAT)   // 512
#define KTOT      (KBASIS + IN_FEAT)      // 576 (self-loop appended)
#define SROW      578                      // padded row stride (bank-conflict avoidance)
#define TILE_M    16
#define BLOCK     128                      // 4 waves (wave32)

// Async global->LDS (GLOBAL_LOAD_ASYNC_TO_LDS_B128, ASYNCcnt-tracked) if declared.
#if defined(__has_builtin)
# if __has_builtin(__builtin_amdgcn_global_load_async_to_lds_b128)
#  define HAVE_ASYNC_LDS_B128 1
# endif
# if __has_builtin(__builtin_amdgcn_s_wait_asynccnt)
#  define HAVE_WAIT_ASYNC 1
# endif
#endif

__global__ __launch_bounds__(BLOCK)
void rgcn_fused_kernel(const float* __restrict__ feat,
                       const float* __restrict__ coeff,
                       const float* __restrict__ W,        // (8,64,64) == (512,64)
                       const float* __restrict__ h_bias,   // (64,)
                       const float* __restrict__ loopw,    // (64,64)
                       const int*   __restrict__ src,
                       const int*   __restrict__ dst,      // sorted ascending
                       const int*   __restrict__ etypes,
                       float*       __restrict__ out,      // (N,64)
                       int n_nodes, int n_edges)
{
    __shared__ float S[TILE_M][SROW];      // 16 x 576 A-matrix (basis accum + self feat)

    const int tid   = threadIdx.x;
    const int lane  = tid & 31;
    const int wave  = tid >> 5;
    const int node0 = blockIdx.x * TILE_M;

    // ---- self-loop features -> LDS K=512..575 (async DMA path when available) ----
#if HAVE_ASYNC_LDS_B128
    for (int c = tid; c < TILE_M * (IN_FEAT / 4); c += BLOCK) {   // 256 x 16B chunks
        int d = c >> 4, q = c & 15;
        int node = node0 + d;
        if (node >= n_nodes) node = n_nodes - 1;                   // safe addr; rows unused
        const float* g = feat + node * IN_FEAT + q * 4;
        __builtin_amdgcn_global_load_async_to_lds_b128(
            (gv4i_p)g, (lv4i_p)&S[d][KBASIS + q * 4],
            /*offset=*/0, /*cpol=*/0);
    }
#else
    for (int c = tid; c < TILE_M * (IN_FEAT / 4); c += BLOCK) {
        int d = c >> 4, q = c & 15;
        int node = node0 + d;
        float4 v = (node < n_nodes) ? ((const float4*)(feat + node * IN_FEAT))[q]
                                    : make_float4(0.f, 0.f, 0.f, 0.f);
        *(float4*)&S[d][KBASIS + q * 4] = v;
    }
#endif

    // ---- zero basis accumulators (overlaps with the async copy above) ----
    for (int i = tid; i < TILE_M * KBASIS; i += BLOCK)
        S[i / KBASIS][i % KBASIS] = 0.0f;

    // ---- edge range [e0,e1) for dst in [node0, node0+16): dst is sorted ----
    int lo = 0, hi = n_edges;
    while (lo < hi) { int mid = (lo + hi) >> 1; if (dst[mid] < node0) lo = mid + 1; else hi = mid; }
    const int e0 = lo;
    hi = n_edges;
    while (lo < hi) { int mid = (lo + hi) >> 1; if (dst[mid] < node0 + TILE_M) lo = mid + 1; else hi = mid; }
    const int e1 = lo;

#if HAVE_ASYNC_LDS_B128
# if HAVE_WAIT_ASYNC
    __builtin_amdgcn_s_wait_asynccnt(0);
# else
    asm volatile("s_wait_asynccnt 0x0" ::: "memory");
# endif
#endif
    __syncthreads();

    // ---- scatter phase: fold relation->basis per edge, accumulate in LDS ----
    // One edge per wave per iteration; lane L covers features 2L, 2L+1.
    for (int e = e0 + wave; e < e1; e += (BLOCK / 32)) {
        const int s = src[e];
        const int r = etypes[e];
        const int d = dst[e] - node0;
        const float2 v = ((const float2*)(feat + s * IN_FEAT))[lane];
        const float* crow = coeff + r * NUM_BASES;
        #pragma unroll
        for (int b = 0; b < NUM_BASES; ++b) {
            const float c = crow[b];
            atomicAdd(&S[d][b * IN_FEAT + 2 * lane],     c * v.x);  // ds_add_f32
            atomicAdd(&S[d][b * IN_FEAT + 2 * lane + 1], c * v.y);
        }
    }
    __syncthreads();

    // ---- GEMM phase: (16 x 576) @ (576 x 64) via V_WMMA_F32_16X16X4_F32 ----
    // Wave w owns output columns [16w, 16w+16). 144 chained WMMAs over K=576.
    const int n0   = wave * 16;
    const int row  = lane & 15;       // M for A, N for B/C/D
    const int half = lane >> 4;       // k-pair select / M+8 select
    const int col  = n0 + row;

    v8f acc;
    const float bias = h_bias[col];
    #pragma unroll
    for (int j = 0; j < 8; ++j) acc[j] = bias;

    for (int k0 = 0; k0 < KTOT; k0 += 4) {
        const int k = k0 + half * 2;
        // A fragment (16x4): lanes0-15 K=k0,k0+1 ; lanes16-31 K=k0+2,k0+3
        v2f a;
        a.x = S[row][k];
        a.y = S[row][k + 1];
        // B fragment (4x16), same K mapping; rows k>=512 come from loop_weight
        const float* wb = (k < KBASIS) ? (W + k * OUT_FEAT)
                                       : (loopw + (k - KBASIS) * OUT_FEAT);
        v2f b;
        b.x = wb[col];
        b.y = wb[OUT_FEAT + col];     // row k+1 (never straddles the 512 boundary: k even)
        acc = __builtin_amdgcn_wmma_f32_16x16x4_f32(
                  /*neg_a=*/false, a, /*neg_b=*/false, b,
                  /*c_mod=*/(short)0, acc, /*reuse_a=*/false, /*reuse_b=*/false);
    }

    // ---- ReLU + store. C/D layout: VGPR j -> M = j + 8*half, N = col ----
    if (node0 + TILE_M <= n_nodes) {
        // full tile: straight-line stores, no per-element exec juggling
        #pragma unroll
        for (int j = 0; j < 8; ++j) {
            const int m = node0 + j + half * 8;
            float v = acc[j];
            out[m * OUT_FEAT + col] = v > 0.0f ? v : 0.0f;
        }
    } else {
        #pragma unroll
        for (int j = 0; j < 8; ++j) {
            const int m = node0 + j + half * 8;
            if (m < n_nodes) {
                float v = acc[j];
                out[m * OUT_FEAT + col] = v > 0.0f ? v : 0.0f;
            }
        }
    }
}

extern "C" void kernel_launch(void* const* d_in, const int* in_sizes, int n_in,
                              void* d_out, int out_size, void* d_ws, size_t ws_size,
                              hipStream_t stream) {
    const float* feat   = (const float*)d_in[0];
    const float* coeff  = (const float*)d_in[1];
    const float* W      = (const float*)d_in[2];
    const float* h_bias = (const float*)d_in[3];
    const float* loopw  = (const float*)d_in[4];
    const int*   src    = (const int*)d_in[5];
    const int*   dst    = (const int*)d_in[6];
    const int*   etyp   = (const int*)d_in[7];
    float*       out    = (float*)d_out;

    const int n_nodes = in_sizes[0] / IN_FEAT;
    const int n_edges = in_sizes[5];
    const int grid    = (n_nodes + TILE_M - 1) / TILE_M;   // 6250 for N=100000

    rgcn_fused_kernel<<<grid, BLOCK, 0, stream>>>(
        feat, coeff, W, h_bias, loopw, src, dst, etyp, out, n_nodes, n_edges);
}